// MNB_455266533601
// MI455X (gfx1250) — compile-verified
//
#include <hip/hip_runtime.h>
#include <cstdint>

typedef __attribute__((ext_vector_type(2))) float v2f;
typedef __attribute__((ext_vector_type(8))) float v8f;

#define VSZ   50000   // vocab size (weight table entries)
#define S_TOK 200     // tokens per phrase
#define B_PHR 2048    // phrases

// One wave (32 lanes) produces 16 phrase outputs via V_WMMA_F32_16X16X4_F32.
// Block = 1024 threads = 32 waves = 512 phrases; grid = 4 blocks.
// Dynamic LDS = 200,000 B holds the full f32 weight table (fits CDNA5's 320KB WGP LDS).
__global__ __launch_bounds__(1024)
void MNB_phrase_linear_kernel(const int* __restrict__ text,    // (S, B) row-major
                              const float* __restrict__ w,     // (V,)
                              const float* __restrict__ bias,  // (1,)
                              float* __restrict__ out)         // (B,)
{
    extern __shared__ float wlds[];
    const int tid = threadIdx.x;

    // ---- Stage the 200KB weight table into LDS via async global->LDS DMA ----
    {
        const float* gsrc = w + tid;
        // Generic pointers to LDS carry the LDS byte offset in their low 32 bits
        // (flat aperture).
        uint32_t lds_byte = (uint32_t)(uintptr_t)(void*)(wlds + tid);
        for (int i = tid; i < VSZ; i += 1024) {
            asm volatile("global_load_async_to_lds_b32 %0, %1, off"
                         :: "v"(lds_byte), "v"(gsrc)
                         : "memory");
            gsrc     += 1024;
            lds_byte += 1024u * 4u;
        }
        asm volatile("s_wait_asynccnt 0" ::: "memory");
    }
    __syncthreads();

    // ---- Per-wave WMMA reduction ----
    const int lane  = tid & 31;
    const int wave  = tid >> 5;
    const int n     = lane & 15;              // phrase column within the wave's tile
    const int kbase = (lane >> 4) << 1;       // lanes 0-15: k=0,1 ; lanes 16-31: k=2,3
    const int b     = blockIdx.x * 512 + wave * 16 + n;

    v2f a;
    a.x = 1.0f;                               // A = 16x4 all-ones: D[m][n] = sum_k B[k][n]
    a.y = 1.0f;

    v8f c = {};                               // f32 accumulator, starts at 0

    const int* tcol = text + b;               // column b of (S, B) token matrix
    #pragma unroll 5
    for (int s0 = 0; s0 < S_TOK; s0 += 4) {
        // B-matrix tile: B[k][n] = w[text[s0+k, b]] ; this lane supplies k = kbase, kbase+1
        const int t0 = tcol[(s0 + kbase) * B_PHR];
        const int t1 = tcol[(s0 + kbase + 1) * B_PHR];
        v2f bm;
        bm.x = wlds[t0];
        bm.y = wlds[t1];
        // (neg_a, A, neg_b, B, c_mod, C, reuse_a, reuse_b)
        c = __builtin_amdgcn_wmma_f32_16x16x4_f32(false, a, false, bm,
                                                  (short)0, c, false, false);
    }

    // D rows are all identical (A all-ones); VGPR0 on lanes 0-15 holds (M=0, N=lane).
    if (lane < 16) {
        out[b] = c[0] + bias[0];
    }
}

extern "C" void kernel_launch(void* const* d_in, const int* in_sizes, int n_in,
                              void* d_out, int out_size, void* d_ws, size_t ws_size,
                              hipStream_t stream) {
    const int*   text = (const int*)d_in[0];    // (200, 2048) tokens
    const float* w    = (const float*)d_in[1];  // (1, 50000) weights
    const float* bias = (const float*)d_in[2];  // (1,)
    float*       out  = (float*)d_out;          // (2048, 1)

    (void)in_sizes; (void)n_in; (void)out_size; (void)d_ws; (void)ws_size;

    const size_t lds_bytes = (size_t)VSZ * sizeof(float); // 200,000 B
    MNB_phrase_linear_kernel<<<4, 1024, lds_bytes, stream>>>(text, w, bias, out);
}